// MokALayer_56023553409594
// MI455X (gfx1250) — compile-verified
//
#include <hip/hip_runtime.h>

typedef __attribute__((ext_vector_type(2))) float v2f;
typedef __attribute__((ext_vector_type(8))) float v8f;

#define D       4096
#define RANK    4
#define SEQ     2048
#define BATCH   4
#define ROWS    (BATCH * SEQ)     // 8192 rows per modality
#define KSPLIT  4
#define KCHUNK  (D / KSPLIT)      // 1024
#define KSTEPS  (KCHUNK / 4)      // 256 wmma steps per wave

// ---------------------------------------------------------------------------
// zero-init of the h accumulator buffer (atomics accumulate into it)
// ---------------------------------------------------------------------------
__global__ void moka_zero(float* __restrict__ p, int n) {
    int i = blockIdx.x * blockDim.x + threadIdx.x;
    if (i < n) p[i] = 0.0f;
}

// ---------------------------------------------------------------------------
// Phase 1: h[mod] = x[mod] @ A[mod]^T  via V_WMMA_F32_16X16X4_F32
//   grid.x = 3 mods * 4 ksplits * 32 tilegroups = 384 blocks, 256 thr (8 waves)
//   each wave: 32 rows (two 16-row WMMA tiles), K-chunk of 1024
// ---------------------------------------------------------------------------
__global__ __launch_bounds__(256)
void moka_proj(const float* __restrict__ xa, const float* __restrict__ xv,
               const float* __restrict__ xt,
               const float* __restrict__ Aa, const float* __restrict__ Av,
               const float* __restrict__ At,
               float* __restrict__ h)
{
    __shared__ float wlds[KCHUNK * RANK];     // weights chunk, [k][r], 16 KB

    const int bid = blockIdx.x;
    const int mod = bid >> 7;                 // /128
    const int rem = bid & 127;
    const int kc  = rem >> 5;                 // K split index 0..3
    const int tg  = rem & 31;                 // tile group 0..31 (256 rows each)

    const float* __restrict__ x  = (mod == 0) ? xa : (mod == 1) ? xv : xt;
    const float* __restrict__ Aw = (mod == 0) ? Aa : (mod == 1) ? Av : At;

    const int tid = threadIdx.x;

    // stage A-weights chunk into LDS transposed: wlds[k*4 + r] = Aw[r, kc*1024 + k]
    for (int i = tid; i < KCHUNK * RANK; i += 256) {
        int r = i >> 10;           // i / KCHUNK
        int k = i & (KCHUNK - 1);
        wlds[k * RANK + r] = Aw[r * D + kc * KCHUNK + k];
    }
    __syncthreads();

    const int  wave = tid >> 5;
    const int  lane = tid & 31;
    const bool lo   = lane < 16;
    const int  n    = lane & 15;              // column within 16-wide tile
    const int  nk   = n & 3;

    const int base = tg * 256 + wave * 32;    // first row of tile0
    const int row0 = base + n;                // this lane's row in tile0
    const int row1 = row0 + 16;               // tile1
    const int kofA = lo ? 0 : 2;              // ISA A-frag: lanes>=16 hold K=2,3

    const float* p0 = x + (size_t)row0 * D + kc * KCHUNK + kofA;
    const float* p1 = x + (size_t)row1 * D + kc * KCHUNK + kofA;

    const int kof0 = lo ? 0 : 2;              // B-frag VGPR0: K=0 / K=2
    const int kof1 = lo ? 1 : 3;              // B-frag VGPR1: K=1 / K=3

    v8f acc0 = {};
    v8f acc1 = {};

    for (int s = 0; s < KSTEPS; ++s) {
        // A fragments land directly in WMMA layout: float2 per lane
        v2f a0 = *(const v2f*)p0;  p0 += 4;
        v2f a1 = *(const v2f*)p1;  p1 += 4;

        // B fragment (4x16 K-major, N<4 real weights, rest 0)
        const int kk = s * 4;
        float w0 = wlds[(kk + kof0) * RANK + nk];
        float w1 = wlds[(kk + kof1) * RANK + nk];
        v2f bf;
        bf.x = (n < RANK) ? w0 : 0.0f;
        bf.y = (n < RANK) ? w1 : 0.0f;

        acc0 = __builtin_amdgcn_wmma_f32_16x16x4_f32(
                   false, a0, false, bf, (short)0, acc0, false, false);
        acc1 = __builtin_amdgcn_wmma_f32_16x16x4_f32(
                   false, a1, false, bf, (short)0, acc1, false, false);
    }

    // D layout: VGPR j -> row (j | j+8), col = lane%16. Keep cols 0..3 only.
    if (n < RANK) {
        const int half = lo ? 0 : 8;
        float* hb = h + (size_t)mod * ROWS * RANK + n;
        #pragma unroll
        for (int j = 0; j < 8; ++j) {
            atomicAdd(hb + (size_t)(base + half + j)      * RANK, acc0[j]);
            atomicAdd(hb + (size_t)(base + 16 + half + j) * RANK, acc1[j]);
        }
    }
}

// ---------------------------------------------------------------------------
// Phase 2: rank-4 cross attention (audio->text, visual->text), flash-style.
//   h_t for one batch = 2048 x 4 f32 = 32 KB -> LDS resident.
//   grid = (qchunk 8, batch 4, modq 2); 1 query per thread, two-pass softmax.
// ---------------------------------------------------------------------------
__global__ __launch_bounds__(256)
void moka_attn(const float* __restrict__ h, float* __restrict__ henh)
{
    __shared__ float4 kt[SEQ];                // 32 KB of keys (h_t)

    const int b    = blockIdx.y;
    const int modq = blockIdx.z;              // 0=audio, 1=visual
    const float4* ht = (const float4*)(h + (size_t)(2 * ROWS + b * SEQ) * RANK);
    const float4* hq = (const float4*)(h + (size_t)(modq * ROWS + b * SEQ) * RANK);

    const int tid = threadIdx.x;
    for (int i = tid; i < SEQ; i += 256) kt[i] = ht[i];
    __syncthreads();

    const int q = blockIdx.x * 256 + tid;
    const float4 qv = hq[q];

    // pass 1: row max of scores (scale = 1/sqrt(R) = 0.5)
    float m = -3.0e38f;
    for (int k = 0; k < SEQ; ++k) {
        float4 kv = kt[k];
        float s = 0.5f * (qv.x * kv.x + qv.y * kv.y + qv.z * kv.z + qv.w * kv.w);
        m = fmaxf(m, s);
    }
    // pass 2: softmax-weighted sum of keys
    float d = 0.0f, ax = 0.0f, ay = 0.0f, az = 0.0f, aw = 0.0f;
    for (int k = 0; k < SEQ; ++k) {
        float4 kv = kt[k];
        float s = 0.5f * (qv.x * kv.x + qv.y * kv.y + qv.z * kv.z + qv.w * kv.w);
        float p = __expf(s - m);
        d += p;
        ax += p * kv.x; ay += p * kv.y; az += p * kv.z; aw += p * kv.w;
    }
    const float inv = 1.0f / d;
    float4 o;                                   // h_enh = h_q + 1.0 * context
    o.x = qv.x + ax * inv;
    o.y = qv.y + ay * inv;
    o.z = qv.z + az * inv;
    o.w = qv.w + aw * inv;
    ((float4*)henh)[(size_t)modq * ROWS + b * SEQ + q] = o;
}

// ---------------------------------------------------------------------------
// Phase 3: out[b, m*2048+s, :] = h_src[b,s,:] @ B_w^T  (rank-4, store-bound)
//   grid = (o-chunk 4 of 1024, seg 12 = b*3+m, s-chunk 16 of 128); 256 thr.
//   each thread owns 4 output columns -> one float4 store per s (fully
//   coalesced 4 KB per block-iteration); B_w rows held in registers.
// ---------------------------------------------------------------------------
__global__ __launch_bounds__(256)
void moka_out(const float* __restrict__ h, const float* __restrict__ henh,
              const float* __restrict__ Bw, float* __restrict__ out)
{
    const int tid = threadIdx.x;
    const int seg = blockIdx.y;               // b*3 + m
    const int b   = seg / 3;
    const int m   = seg % 3;

    const float4* hs = (const float4*)((m == 2)
        ? (h    + (size_t)(2 * ROWS + b * SEQ) * RANK)
        : (henh + (size_t)(m * ROWS + b * SEQ) * RANK));

    const int o = blockIdx.x * 1024 + tid * 4;
    const float4* Bw4 = (const float4*)Bw;    // row o of B_w = Bw4[o]
    const float4 w0 = Bw4[o + 0];
    const float4 w1 = Bw4[o + 1];
    const float4 w2 = Bw4[o + 2];
    const float4 w3 = Bw4[o + 3];

    float* ob = out + (size_t)seg * SEQ * 4096 + o;
    const int s0 = blockIdx.z * 128;
    for (int s = s0; s < s0 + 128; ++s) {
        float4 hv = hs[s];                    // uniform -> broadcast load
        float4 r;
        r.x = hv.x * w0.x + hv.y * w0.y + hv.z * w0.z + hv.w * w0.w;
        r.y = hv.x * w1.x + hv.y * w1.y + hv.z * w1.z + hv.w * w1.w;
        r.z = hv.x * w2.x + hv.y * w2.y + hv.z * w2.z + hv.w * w2.w;
        r.w = hv.x * w3.x + hv.y * w3.y + hv.z * w3.z + hv.w * w3.w;
        *(float4*)(ob + (size_t)s * 4096) = r;
    }
}

// ---------------------------------------------------------------------------
extern "C" void kernel_launch(void* const* d_in, const int* in_sizes, int n_in,
                              void* d_out, int out_size, void* d_ws, size_t ws_size,
                              hipStream_t stream)
{
    const float* xa = (const float*)d_in[0];
    const float* xv = (const float*)d_in[1];
    const float* xt = (const float*)d_in[2];
    const float* Aa = (const float*)d_in[3];
    const float* Av = (const float*)d_in[4];
    const float* At = (const float*)d_in[5];
    const float* Bw = (const float*)d_in[6];

    float* h    = (float*)d_ws;                       // 3 * 8192 * 4 floats
    float* henh = h + (size_t)3 * ROWS * RANK;        // 2 * 8192 * 4 floats

    const int hElems = 3 * ROWS * RANK;               // 98304
    moka_zero<<<(hElems + 255) / 256, 256, 0, stream>>>(h, hElems);

    moka_proj<<<384, 256, 0, stream>>>(xa, xv, xt, Aa, Av, At, h);

    moka_attn<<<dim3(8, BATCH, 2), 256, 0, stream>>>(h, henh);

    moka_out<<<dim3(4, 12, 16), 256, 0, stream>>>(h, henh, Bw, (float*)d_out);
}